// Encoder_63522566308145
// MI455X (gfx1250) — compile-verified
//
#include <hip/hip_runtime.h>
#include <hip/hip_bf16.h>

typedef __attribute__((ext_vector_type(2))) float v2f;
typedef __attribute__((ext_vector_type(8))) float v8f;

// ---------------------------------------------------------------------------
// Degree: deg[dst] += 1 per edge (self-loops folded into dinv kernel below)
// ---------------------------------------------------------------------------
__global__ void deg_kernel(const int* __restrict__ dst, float* __restrict__ deg,
                           int E) {
    int e = blockIdx.x * blockDim.x + threadIdx.x;
    if (e < E) atomicAdd(&deg[dst[e]], 1.0f);
}

// dinv[i] = rsqrt(deg[i] + 1)   (+1 = self loop; always > 0)
__global__ void dinv_kernel(float* __restrict__ deg, int n) {
    int i = blockIdx.x * blockDim.x + threadIdx.x;
    if (i < n) deg[i] = __frsqrt_rn(deg[i] + 1.0f);
}

// ---------------------------------------------------------------------------
// GEMM1: p1[n,64] = x[n,128] @ W1[128,64], fp32 WMMA 16x16x4.
// blockDim = 128 (4 waves); wave w handles column tile n0 = 16*w; grid = n/16.
// A-frag (16x4): lanes 0-15 carry K = k,k+1 ; lanes 16-31 carry K = k+2,k+3.
// B-frag (4x16): same K split across lane halves; N = lane&15.
// C/D: VGPR r -> M = m0 + (lane>=16?8:0) + r, N = n0 + (lane&15).
// ---------------------------------------------------------------------------
__global__ __launch_bounds__(128) void gemm1_kernel(
    const float* __restrict__ x, const float* __restrict__ W,
    float* __restrict__ p1) {
    const int wave  = threadIdx.x >> 5;
    const int lane  = threadIdx.x & 31;
    const int l15   = lane & 15;
    const int khalf = (lane >> 4) << 1;     // 0 or 2
    const int m0    = blockIdx.x << 4;
    const int n0    = wave << 4;

    const float* __restrict__ xrow = x + (long long)(m0 + l15) * 128;
    v8f c = {};
    #pragma unroll 4
    for (int k = 0; k < 128; k += 4) {
        v2f a, b;
        a.x = xrow[k + khalf];
        a.y = xrow[k + khalf + 1];
        b.x = W[(k + khalf) * 64 + n0 + l15];
        b.y = W[(k + khalf + 1) * 64 + n0 + l15];
        c = __builtin_amdgcn_wmma_f32_16x16x4_f32(
                false, a, false, b, (short)0, c, false, false);
    }
    const int col   = n0 + l15;
    const int rbase = m0 + ((lane >> 4) << 3);
    #pragma unroll
    for (int r = 0; r < 8; ++r)
        p1[(long long)(rbase + r) * 64 + col] = c[r];
}

// ---------------------------------------------------------------------------
// Scatter for conv1: 64 threads per edge, coalesced gather + fp32 atomics.
// ---------------------------------------------------------------------------
__global__ void scatter64_kernel(const float* __restrict__ p1,
                                 const float* __restrict__ dinv,
                                 const int* __restrict__ src,
                                 const int* __restrict__ dst,
                                 float* __restrict__ a1, int E) {
    long long t = (long long)blockIdx.x * blockDim.x + threadIdx.x;
    long long e = t >> 6;
    int j = (int)(t & 63);
    if (e >= E) return;
    int s = src[e], d = dst[e];
    float w = dinv[s] * dinv[d];
    atomicAdd(&a1[(long long)d * 64 + j], p1[(long long)s * 64 + j] * w);
}

// Epilogue conv1: add self-loop term + bias, ReLU, in place -> h (== a1 buf)
__global__ void epilogue1_kernel(float* __restrict__ a1,
                                 const float* __restrict__ p1,
                                 const float* __restrict__ dinv,
                                 const float* __restrict__ b1, int n) {
    long long i = (long long)blockIdx.x * blockDim.x + threadIdx.x;
    if (i >= (long long)n * 64) return;
    int node = (int)(i >> 6), j = (int)(i & 63);
    float di = dinv[node];
    float v = a1[i] + p1[i] * di * di + b1[j];
    a1[i] = v > 0.0f ? v : 0.0f;
}

// ---------------------------------------------------------------------------
// Fused GEMMs: pmu = h@W_mu, plog = h@W_log  ([n,64]@[64,32]).
// 4 waves: wave 0/1 -> W_mu tiles n0=0,16 ; wave 2/3 -> W_log tiles.
// ---------------------------------------------------------------------------
__global__ __launch_bounds__(128) void gemm23_kernel(
    const float* __restrict__ h, const float* __restrict__ Wmu,
    const float* __restrict__ Wlog, float* __restrict__ pmu,
    float* __restrict__ plog) {
    const int wave  = threadIdx.x >> 5;
    const int lane  = threadIdx.x & 31;
    const int l15   = lane & 15;
    const int khalf = (lane >> 4) << 1;
    const int m0    = blockIdx.x << 4;
    const int n0    = (wave & 1) << 4;
    const float* __restrict__ W = (wave < 2) ? Wmu : Wlog;
    float* __restrict__ p       = (wave < 2) ? pmu : plog;

    const float* __restrict__ hrow = h + (long long)(m0 + l15) * 64;
    v8f c = {};
    #pragma unroll 4
    for (int k = 0; k < 64; k += 4) {
        v2f a, b;
        a.x = hrow[k + khalf];
        a.y = hrow[k + khalf + 1];
        b.x = W[(k + khalf) * 32 + n0 + l15];
        b.y = W[(k + khalf + 1) * 32 + n0 + l15];
        c = __builtin_amdgcn_wmma_f32_16x16x4_f32(
                false, a, false, b, (short)0, c, false, false);
    }
    const int col   = n0 + l15;
    const int rbase = m0 + ((lane >> 4) << 3);
    #pragma unroll
    for (int r = 0; r < 8; ++r)
        p[(long long)(rbase + r) * 32 + col] = c[r];
}

// ---------------------------------------------------------------------------
// Scatter for conv2/3: 64 threads per edge; j<32 -> mu, j>=32 -> logvar.
// ---------------------------------------------------------------------------
__global__ void scatter32x2_kernel(const float* __restrict__ pmu,
                                   const float* __restrict__ plog,
                                   const float* __restrict__ dinv,
                                   const int* __restrict__ src,
                                   const int* __restrict__ dst,
                                   float* __restrict__ out_mu,
                                   float* __restrict__ out_log, int E) {
    long long t = (long long)blockIdx.x * blockDim.x + threadIdx.x;
    long long e = t >> 6;
    int j = (int)(t & 63);
    if (e >= E) return;
    int s = src[e], d = dst[e];
    float w = dinv[s] * dinv[d];
    if (j < 32)
        atomicAdd(&out_mu[(long long)d * 32 + j],
                  pmu[(long long)s * 32 + j] * w);
    else
        atomicAdd(&out_log[(long long)d * 32 + (j - 32)],
                  plog[(long long)s * 32 + (j - 32)] * w);
}

// Epilogue conv2/3: add self-loop term + bias for both outputs.
__global__ void epilogue2_kernel(float* __restrict__ out_mu,
                                 float* __restrict__ out_log,
                                 const float* __restrict__ pmu,
                                 const float* __restrict__ plog,
                                 const float* __restrict__ dinv,
                                 const float* __restrict__ bmu,
                                 const float* __restrict__ blog, int n) {
    long long i = (long long)blockIdx.x * blockDim.x + threadIdx.x;
    if (i >= (long long)n * 32) return;
    int node = (int)(i >> 5), j = (int)(i & 31);
    float di = dinv[node];
    float d2 = di * di;
    out_mu[i]  += pmu[i]  * d2 + bmu[j];
    out_log[i] += plog[i] * d2 + blog[j];
}

// ---------------------------------------------------------------------------
extern "C" void kernel_launch(void* const* d_in, const int* in_sizes, int n_in,
                              void* d_out, int out_size, void* d_ws,
                              size_t ws_size, hipStream_t stream) {
    const float* x    = (const float*)d_in[0];   // [n,128]
    const int*   ei   = (const int*)d_in[1];     // [2,E]
    const float* W1   = (const float*)d_in[2];   // [128,64]
    const float* b1   = (const float*)d_in[3];   // [64]
    const float* Wmu  = (const float*)d_in[4];   // [64,32]
    const float* bmu  = (const float*)d_in[5];   // [32]
    const float* Wlog = (const float*)d_in[6];   // [64,32]
    const float* blog = (const float*)d_in[7];   // [32]

    const int n = in_sizes[0] / 128;             // 100000
    const int E = in_sizes[1] / 2;               // 1600000
    const int* src = ei;
    const int* dst = ei + E;

    // Workspace layout (floats): dinv[n] | p1[n*64] | a1[n*64]
    float* dinv = (float*)d_ws;
    float* p1   = dinv + n;
    float* a1   = p1 + (long long)n * 64;
    // After conv1, p1 is dead -> reuse for projections of h.
    float* pmu  = p1;
    float* plog = p1 + (long long)n * 32;

    float* out_mu  = (float*)d_out;
    float* out_log = out_mu + (long long)n * 32;

    // Zero accumulators (graph-capture-safe async memsets).
    hipMemsetAsync(dinv, 0, (size_t)n * sizeof(float), stream);
    hipMemsetAsync(a1, 0, (size_t)n * 64 * sizeof(float), stream);
    hipMemsetAsync(d_out, 0, (size_t)out_size * sizeof(float), stream);

    // Degree + normalization.
    deg_kernel<<<(E + 255) / 256, 256, 0, stream>>>(dst, dinv, E);
    dinv_kernel<<<(n + 255) / 256, 256, 0, stream>>>(dinv, n);

    // conv1
    gemm1_kernel<<<n / 16, 128, 0, stream>>>(x, W1, p1);
    {
        long long work = (long long)E * 64;
        scatter64_kernel<<<(int)((work + 255) / 256), 256, 0, stream>>>(
            p1, dinv, src, dst, a1, E);
    }
    {
        long long work = (long long)n * 64;
        epilogue1_kernel<<<(int)((work + 255) / 256), 256, 0, stream>>>(
            a1, p1, dinv, b1, n);
    }

    // conv2 + conv3 (fused projections; h lives in a1)
    gemm23_kernel<<<n / 16, 128, 0, stream>>>(a1, Wmu, Wlog, pmu, plog);
    {
        long long work = (long long)E * 64;
        scatter32x2_kernel<<<(int)((work + 255) / 256), 256, 0, stream>>>(
            pmu, plog, dinv, src, dst, out_mu, out_log, E);
    }
    {
        long long work = (long long)n * 32;
        epilogue2_kernel<<<(int)((work + 255) / 256), 256, 0, stream>>>(
            out_mu, out_log, pmu, plog, dinv, bmu, blog, n);
    }
}